// DDI_GraphTransformer_5351529251074
// MI455X (gfx1250) — compile-verified
//
#include <hip/hip_runtime.h>
#include <hip/hip_bf16.h>
#include <math.h>

typedef __attribute__((ext_vector_type(2))) float v2f;
typedef __attribute__((ext_vector_type(8))) float v8f;

#define HEADS 8
#define CH 16
#define DHID 128   // HEADS*CH

// ---- ordered-uint encoding of floats so atomicMax(u32) == float max ----
__device__ __forceinline__ unsigned f2ord(float f) {
    unsigned u = __float_as_uint(f);
    return (u & 0x80000000u) ? ~u : (u | 0x80000000u);
}
__device__ __forceinline__ float ord2f(unsigned o) {
    unsigned u = (o & 0x80000000u) ? (o ^ 0x80000000u) : ~o;
    return __uint_as_float(u);
}

// ---- zero/neg-inf init of per-layer accumulators (ws is poisoned 0xAA) ----
__global__ void ddi_init(unsigned* __restrict__ mord, float* __restrict__ denom,
                         float* __restrict__ acc, int N) {
    int idx = blockIdx.x * blockDim.x + threadIdx.x;
    if (idx < N * HEADS) { mord[idx] = 0u; denom[idx] = 0.0f; }
    if (idx < N * DHID)  acc[idx] = 0.0f;
}

// ---- fused Q/K/V/skip projections: [N,128] @ [128,128] + bias, fp32 WMMA ----
// grid = ((N+15)/16, 4); block = 256 (8 wave32). Wave w -> output col tile w.
// blockIdx.y selects which weight matrix / destination.
__global__ __launch_bounds__(256) void ddi_gemm_qkvs(
    int N, const float* __restrict__ X,
    const float* __restrict__ Wq, const float* __restrict__ Bq,
    const float* __restrict__ Wk, const float* __restrict__ Bk,
    const float* __restrict__ Wv, const float* __restrict__ Bv,
    const float* __restrict__ Ws, const float* __restrict__ Bs,
    float* __restrict__ Q, float* __restrict__ K,
    float* __restrict__ V, float* __restrict__ S)
{
    const int rowTile = blockIdx.x;
    const int wave = threadIdx.x >> 5;   // 0..7 -> column tile
    const int lane = threadIdx.x & 31;
    const int m  = lane & 15;            // row (A) / col (B,D) within tile
    const int hi = lane >> 4;            // K-half selector for A/B frags

    const float* W; const float* B; float* O;
    switch (blockIdx.y) {
        case 0:  W = Wq; B = Bq; O = Q; break;
        case 1:  W = Wk; B = Bk; O = K; break;
        case 2:  W = Wv; B = Bv; O = V; break;
        default: W = Ws; B = Bs; O = S; break;
    }

    const int row = rowTile * 16 + m;
    const int rl  = row < N ? row : 0;       // clamp; garbage rows never stored
    const int col = wave * 16 + m;
    const float* xrow = X + (size_t)rl * DHID;

    v8f acc = {};
#pragma unroll
    for (int k = 0; k < DHID; k += 4) {
        const int kk = k + hi * 2;
        v2f a, b;
        // A 16x4 layout: lane m holds row m; VGPR0/1 = K={kk,kk+1}, halves by lane>=16
        a.x = xrow[kk];
        a.y = xrow[kk + 1];
        // B 4x16 layout mirrored: lane m holds col m; K halves split by lane>=16
        b.x = W[(size_t)kk * DHID + col];
        b.y = W[(size_t)(kk + 1) * DHID + col];
        acc = __builtin_amdgcn_wmma_f32_16x16x4_f32(false, a, false, b,
                                                    (short)0, acc, false, false);
    }
    const float bb = B[col];
    // D layout: lane gives col; VGPR v gives row v + 8*(lane>=16)
#pragma unroll
    for (int v = 0; v < 8; ++v) {
        const int r = rowTile * 16 + v + 8 * hi;
        if (r < N) O[(size_t)r * DHID + col] = acc[v] + bb;
    }
}

// ---- pass A: per-(edge,head) logit + segment max (ordered-uint atomicMax) ----
__global__ void ddi_edge_logits(const int* __restrict__ ei, int E,
                                const float* __restrict__ Q, const float* __restrict__ K,
                                float* __restrict__ alpha, unsigned* __restrict__ mord)
{
    int idx = blockIdx.x * blockDim.x + threadIdx.x;
    if (idx >= E * HEADS) return;
    const int e = idx >> 3, h = idx & 7;
    const int s = ei[e];        // src
    const int d = ei[E + e];    // dst
    const float4* qp = (const float4*)(Q + (size_t)d * DHID + h * CH);
    const float4* kp = (const float4*)(K + (size_t)s * DHID + h * CH);
    float sum = 0.f;
#pragma unroll
    for (int i = 0; i < 4; ++i) {
        float4 a = qp[i], b = kp[i];
        sum += a.x * b.x + a.y * b.y + a.z * b.z + a.w * b.w;
    }
    sum *= 0.25f;               // 1/sqrt(CH)
    alpha[idx] = sum;
    atomicMax(&mord[d * HEADS + h], f2ord(sum));
}

// ---- pass B: a = exp(alpha - max); denom += a; acc[dst] += a * v[src] ----
__global__ void ddi_edge_agg(const int* __restrict__ ei, int E,
                             const float* __restrict__ V, const float* __restrict__ alpha,
                             const unsigned* __restrict__ mord,
                             float* __restrict__ denom, float* __restrict__ acc)
{
    int idx = blockIdx.x * blockDim.x + threadIdx.x;
    if (idx >= E * HEADS) return;
    const int e = idx >> 3, h = idx & 7;
    const int s = ei[e];
    const int d = ei[E + e];
    const float mx = ord2f(mord[d * HEADS + h]);
    const float a = expf(alpha[idx] - mx);
    unsafeAtomicAdd(&denom[d * HEADS + h], a);
    const float4* vp = (const float4*)(V + (size_t)s * DHID + h * CH);
    float* ap = acc + (size_t)d * DHID + h * CH;
#pragma unroll
    for (int i = 0; i < 4; ++i) {
        float4 vv = vp[i];
        unsafeAtomicAdd(&ap[4 * i + 0], a * vv.x);
        unsafeAtomicAdd(&ap[4 * i + 1], a * vv.y);
        unsafeAtomicAdd(&ap[4 * i + 2], a * vv.z);
        unsafeAtomicAdd(&ap[4 * i + 3], a * vv.w);
    }
}

// ---- node pass: out = acc/denom + skip (optional ReLU) ----
__global__ void ddi_finalize(int N, const float* __restrict__ acc,
                             const float* __restrict__ denom, const float* __restrict__ S,
                             float* __restrict__ out, int do_relu)
{
    int idx = blockIdx.x * blockDim.x + threadIdx.x;
    if (idx >= N * DHID) return;
    const int i = idx >> 7;
    const int d = idx & 127;
    const int h = d >> 4;
    const float den = denom[i * HEADS + h];
    float o = (den > 0.f ? acc[idx] / den : 0.f) + S[idx];
    if (do_relu) o = fmaxf(o, 0.f);
    out[idx] = o;
}

extern "C" void kernel_launch(void* const* d_in, const int* in_sizes, int n_in,
                              void* d_out, int out_size, void* d_ws, size_t ws_size,
                              hipStream_t stream) {
    const float* x  = (const float*)d_in[0];
    const int*   ei = (const int*)d_in[1];    // [2,E]: row0=src, row1=dst
    // d_in[2] edge_attr: unused by the reference forward
    const int N = in_sizes[0] / DHID;
    const int E = in_sizes[1] / 2;

    // workspace carve-up (fp32 unless noted)
    char* p = (char*)d_ws;
    const size_t nodeMat = (size_t)N * DHID * sizeof(float);
    float*    Q     = (float*)p;            p += nodeMat;
    float*    K     = (float*)p;            p += nodeMat;
    float*    V     = (float*)p;            p += nodeMat;
    float*    S     = (float*)p;            p += nodeMat;
    float*    acc   = (float*)p;            p += nodeMat;
    float*    hbuf  = (float*)p;            p += nodeMat;
    float*    alpha = (float*)p;            p += (size_t)E * HEADS * sizeof(float);
    unsigned* mord  = (unsigned*)p;         p += (size_t)N * HEADS * sizeof(unsigned);
    float*    denom = (float*)p;            /* p += N*HEADS*4 */

    const dim3 gemmGrid((N + 15) / 16, 4, 1);
    const int nodeBlocks = (N * DHID + 255) / 256;
    const int edgeBlocks = (E * HEADS + 255) / 256;

    auto run_layer = [&](const float* in,
                         const float* qw, const float* qb,
                         const float* kw, const float* kb,
                         const float* vw, const float* vb,
                         const float* sw, const float* sb,
                         float* out, int do_relu) {
        ddi_init<<<nodeBlocks, 256, 0, stream>>>(mord, denom, acc, N);
        ddi_gemm_qkvs<<<gemmGrid, 256, 0, stream>>>(N, in,
            qw, qb, kw, kb, vw, vb, sw, sb, Q, K, V, S);
        ddi_edge_logits<<<edgeBlocks, 256, 0, stream>>>(ei, E, Q, K, alpha, mord);
        ddi_edge_agg<<<edgeBlocks, 256, 0, stream>>>(ei, E, V, alpha, mord, denom, acc);
        ddi_finalize<<<nodeBlocks, 256, 0, stream>>>(N, acc, denom, S, out, do_relu);
    };

    // layer 1: x -> hbuf (ReLU)
    run_layer(x,
              (const float*)d_in[3],  (const float*)d_in[4],
              (const float*)d_in[5],  (const float*)d_in[6],
              (const float*)d_in[7],  (const float*)d_in[8],
              (const float*)d_in[9],  (const float*)d_in[10],
              hbuf, 1);
    // layer 2: hbuf -> d_out (no ReLU)
    run_layer(hbuf,
              (const float*)d_in[11], (const float*)d_in[12],
              (const float*)d_in[13], (const float*)d_in[14],
              (const float*)d_in[15], (const float*)d_in[16],
              (const float*)d_in[17], (const float*)d_in[18],
              (float*)d_out, 0);
}